// Edge_Discriminator_22230750724356
// MI455X (gfx1250) — compile-verified
//
#include <hip/hip_runtime.h>
#include <math.h>

// ---------------- problem constants (match reference) ----------------
#define NNODES 6000
#define NEDGES 192000
#define DIN    512
#define HDIM   128
#define EOSV   1e-10f
#define ALPHAV 1.0f
#define TEMPV  1.0f

typedef __attribute__((ext_vector_type(2))) float v2f;
typedef __attribute__((ext_vector_type(4))) float v4f;
typedef __attribute__((ext_vector_type(8))) float v8f;

static constexpr long long NN      = (long long)NNODES * NNODES;
static constexpr long long OFF_LP  = 0;                 // adj_lp   [N*N]
static constexpr long long OFF_HP  = NN;                // adj_hp   [N*N]
static constexpr long long OFF_WLP = 2 * NN;            // weights_lp [E]
static constexpr long long OFF_WHP = 2 * NN + NEDGES;   // weights_hp [E]
static constexpr long long OFF_UN  = 2 * NN + 2 * (long long)NEDGES; // adj_unnorm [N*N]

// ---------------- 1) init scratch: g=0, rowsum d=1 (the +eye term) ----
__global__ void k_init(float* __restrict__ g, float* __restrict__ d_lp,
                       float* __restrict__ d_hp) {
    int i = blockIdx.x * blockDim.x + threadIdx.x;
    if (i < NNODES) { g[i] = 0.0f; d_lp[i] = 1.0f; d_hp[i] = 1.0f; }
}

// ---------------- 2) fused GEMM + ReLU + row-dot via fp32 WMMA --------
// Each block: 16 nodes. 8 waves; wave w owns H columns [16w,16w+16).
// K loop over DIN in steps of 4 using V_WMMA_F32_16X16X4_F32.
// A (16x4 fp32) lane layout: M = lane&15; VGPR0 holds K = 2*(lane>>4),
// VGPR1 holds K+1.  B (4x16 fp32): row striped across lanes:
// VGPR0 = B[K=2*half][N=lane&15], VGPR1 = B[K+1][N].
// C (16x16 f32): VGPR r, lanes 0-15 -> M=r, lanes 16-31 -> M=r+8, N=lane&15.
__global__ __launch_bounds__(256) void k_gemm_g(
    const float* __restrict__ F, const float* __restrict__ Wemb,
    const float* __restrict__ Bemb, const float* __restrict__ Wmlp,
    float* __restrict__ g) {
    const int node0 = blockIdx.x * 16;
    const int wave  = threadIdx.x >> 5;   // 0..7
    const int lane  = threadIdx.x & 31;
    const int half  = lane >> 4;          // 0/1
    const int l16   = lane & 15;
    const int col0  = wave * 16;

    v8f acc = {};
    const float* Arow = F + (long long)(node0 + l16) * DIN;
    #pragma unroll 4
    for (int k = 0; k < DIN; k += 4) {
        const int ka = k + 2 * half;
        v2f a, b;
        a.x = Arow[ka];
        a.y = Arow[ka + 1];
        b.x = Wemb[ka * HDIM + col0 + l16];
        b.y = Wemb[(ka + 1) * HDIM + col0 + l16];
        acc = __builtin_amdgcn_wmma_f32_16x16x4_f32(
            false, a, false, b, (short)0, acc, false, false);
    }

    // bias -> ReLU -> multiply by (wA+wB)[h] -> reduce over the 16 columns
    const int   h    = col0 + l16;
    const float wsum = Wmlp[h] + Wmlp[HDIM + h];
    const float bh   = Bemb[h];
    #pragma unroll
    for (int r = 0; r < 8; ++r) {
        float v = acc[r] + bh;
        v = v > 0.0f ? v : 0.0f;
        v *= wsum;
        // butterfly reduce within each 16-lane half (same node, 16 h-cols)
        v += __shfl_xor(v, 1, 32);
        v += __shfl_xor(v, 2, 32);
        v += __shfl_xor(v, 4, 32);
        v += __shfl_xor(v, 8, 32);
        if (l16 == 0)
            atomicAdd(&g[node0 + r + 8 * half], v);
    }
}

// ---------------- 3) per-edge gating + row-sum accumulation ----------
__global__ void k_edge(const int* __restrict__ edges, const float* __restrict__ g,
                       const float* __restrict__ noise, const float* __restrict__ bmlp,
                       float* __restrict__ out, float* __restrict__ d_lp,
                       float* __restrict__ d_hp) {
    int e = blockIdx.x * blockDim.x + threadIdx.x;
    if (e >= NEDGES) return;
    const int u = edges[e];
    const int v = edges[NEDGES + e];
    const float raw = 0.5f * (g[u] + g[v]) + bmlp[0];
    const float x   = (noise[e] + raw) / TEMPV;
    const float wlp = 1.0f / (1.0f + expf(-x));
    const float whp = 1.0f - wlp;
    out[OFF_WLP + e] = wlp;      // regular-temporal: re-read by k_scatter
    out[OFF_WHP + e] = whp;
    atomicAdd(&d_lp[u], wlp);    // adj.sum(axis=1): row index is edges[0]
    atomicAdd(&d_hp[u], whp);
}

// ---------------- 4) inverse sqrt degree ------------------------------
__global__ void k_inv(const float* __restrict__ d_lp, const float* __restrict__ d_hp,
                      float* __restrict__ inv_lp, float* __restrict__ inv_hp) {
    int i = blockIdx.x * blockDim.x + threadIdx.x;
    if (i < NNODES) {
        inv_lp[i] = 1.0f / (sqrtf(d_lp[i]) + EOSV);
        inv_hp[i] = 1.0f / (sqrtf(d_hp[i]) + EOSV);
    }
}

// ---------------- 5) single-pass dense fill (v4f, NT stores) ----------
// 432 MB streamed once and never re-read: non-temporal so the 192MB L2
// isn't thrashed.  adj_lp = 0 off-diag, inv_i^2 on diag; adj_hp = eye;
// adj_unnorm = 0.
__global__ __launch_bounds__(256) void k_fill(float* __restrict__ out,
                                              const float* __restrict__ inv_lp) {
    const int ROW_SLOTS = 1536;               // 6 blocks * 256 threads (guard 1500)
    long long idx  = (long long)blockIdx.x * blockDim.x + threadIdx.x;
    long long row  = idx / ROW_SLOTS;
    int       slot = (int)(idx % ROW_SLOTS);
    if (slot >= NNODES / 4 || row >= NNODES) return;
    const int  c    = slot * 4;
    long long  base = row * NNODES + c;
    v4f z  = {0.f, 0.f, 0.f, 0.f};
    v4f lp = z, hp = z;
    const int r = (int)row;
    if (r >= c && r < c + 4) {
        const float il = inv_lp[r];
        lp[r - c] = il * il;                  // (A_diag=1) * inv_r * inv_r
        hp[r - c] = 1.0f;                     // eye
    }
    __builtin_nontemporal_store(lp, (v4f*)(out + OFF_LP + base));
    __builtin_nontemporal_store(hp, (v4f*)(out + OFF_HP + base));
    __builtin_nontemporal_store(z,  (v4f*)(out + OFF_UN + base));
}

// ---------------- 6) edge scatter of final normalized values ----------
__global__ void k_scatter(const int* __restrict__ edges, float* __restrict__ out,
                          const float* __restrict__ inv_lp,
                          const float* __restrict__ inv_hp) {
    int e = blockIdx.x * blockDim.x + threadIdx.x;
    if (e >= NEDGES) return;
    const int u = edges[e];
    const int v = edges[NEDGES + e];
    const float wlp = out[OFF_WLP + e];
    const float whp = out[OFF_WHP + e];
    const float sl  = inv_lp[u] * inv_lp[v];
    const float sh  = inv_hp[u] * inv_hp[v];
    float fl, fh;
    if (u == v) {   // self-edge: scatter then +eye before normalization
        fl = (wlp + 1.0f) * sl;
        fh = 1.0f - (whp + 1.0f) * sh * ALPHAV;
    } else {
        fl = wlp * sl;
        fh = -(whp * sh) * ALPHAV;
    }
    const long long p = (long long)u * NNODES + v;
    __builtin_nontemporal_store(fl,   out + OFF_LP + p);
    __builtin_nontemporal_store(fh,   out + OFF_HP + p);
    __builtin_nontemporal_store(1.0f, out + OFF_UN + p);
}

// ---------------- launcher -------------------------------------------
extern "C" void kernel_launch(void* const* d_in, const int* in_sizes, int n_in,
                              void* d_out, int out_size, void* d_ws, size_t ws_size,
                              hipStream_t stream) {
    const float* features   = (const float*)d_in[0];
    const int*   edges      = (const int*)  d_in[1];
    const float* w_emb      = (const float*)d_in[2];
    const float* b_emb      = (const float*)d_in[3];
    const float* w_mlp      = (const float*)d_in[4];
    const float* b_mlp      = (const float*)d_in[5];
    const float* gate_noise = (const float*)d_in[6];
    float* out = (float*)d_out;

    float* g      = (float*)d_ws;
    float* d_lp   = g      + NNODES;
    float* d_hp   = d_lp   + NNODES;
    float* inv_lp = d_hp   + NNODES;
    float* inv_hp = inv_lp + NNODES;

    k_init<<<(NNODES + 255) / 256, 256, 0, stream>>>(g, d_lp, d_hp);
    k_gemm_g<<<NNODES / 16, 256, 0, stream>>>(features, w_emb, b_emb, w_mlp, g);
    k_edge<<<(NEDGES + 255) / 256, 256, 0, stream>>>(edges, g, gate_noise, b_mlp,
                                                     out, d_lp, d_hp);
    k_inv<<<(NNODES + 255) / 256, 256, 0, stream>>>(d_lp, d_hp, inv_lp, inv_hp);
    {
        long long total = (long long)NNODES * 1536;   // 1536 slots per row
        int blocks = (int)((total + 255) / 256);
        k_fill<<<blocks, 256, 0, stream>>>(out, inv_lp);
    }
    k_scatter<<<(NEDGES + 255) / 256, 256, 0, stream>>>(edges, out, inv_lp, inv_hp);
}